// SoftMoE_LLaMa_FFN_46523085750476
// MI455X (gfx1250) — compile-verified
//
#include <hip/hip_runtime.h>
#include <math.h>

// Problem constants
#define B_ 4
#define T_ 8192
#define D_ 1024
#define F_ 4096
#define E_ 8

typedef __attribute__((ext_vector_type(2))) float v2f;
typedef __attribute__((ext_vector_type(8))) float v8f;

// Workspace layout (floats)
constexpr size_t OFF_LOGITS = 0;                                   // B*T*E
constexpr size_t OFF_MAXV   = (size_t)B_ * T_ * E_;                // B*E (padded)
constexpr size_t OFF_RSUM   = OFF_MAXV + 64;                       // B*E (padded)
constexpr size_t OFF_SLOTS  = OFF_RSUM + 64;                       // B*E*D
constexpr size_t OFF_H      = OFF_SLOTS + (size_t)B_ * E_ * D_;    // B*E*F
constexpr size_t OFF_Y      = OFF_H + (size_t)B_ * E_ * F_;        // B*E*D
constexpr size_t N_ZERO     = (size_t)B_ * E_ * D_ + (size_t)B_ * E_ * F_ + (size_t)B_ * E_ * D_;

static __device__ __forceinline__ v8f wmma4(v2f a, v2f b, v8f c) {
  // D = A(16x4,f32) * B(4x16,f32) + C(16x16,f32)
  return __builtin_amdgcn_wmma_f32_16x16x4_f32(false, a, false, b, (short)0, c,
                                               false, false);
}

// ---------------------------------------------------------------------------
// Kernel 0: zero slots / h / y (atomic accumulation targets)
// ---------------------------------------------------------------------------
__global__ __launch_bounds__(256) void k_init(float* __restrict__ ws) {
  size_t i = (size_t)blockIdx.x * 256 + threadIdx.x;
  if (i < N_ZERO) ws[OFF_SLOTS + i] = 0.0f;
}

// ---------------------------------------------------------------------------
// Kernel 1: logits[b,t,e] = X[b,t,:] . slots_w[e,:]   (WMMA, M=T tile, N=E pad16)
// grid: 256 blocks x 256 thr (8 waves/blk, 1 wave = one 16-token tile)
// Pad lanes (e>=8) read a shared all-zero LDS row -> no selects in K loop.
// ---------------------------------------------------------------------------
__global__ __launch_bounds__(256) void k_logits(const float* __restrict__ X,
                                                const float* __restrict__ SW,
                                                float* __restrict__ ws) {
  __shared__ float sw[9 * 1028];  // rows 0-7 = slots_w, row 8 = zeros
  float* logits = ws + OFF_LOGITS;

  for (int i = threadIdx.x; i < E_ * D_; i += 256) {
    int e = i >> 10, k = i & 1023;
    sw[e * 1028 + k] = SW[i];
  }
  for (int i = threadIdx.x; i < 1028; i += 256) sw[8 * 1028 + i] = 0.0f;
  __syncthreads();

  int wid = threadIdx.x >> 5, lane = threadIdx.x & 31;
  int tile = blockIdx.x * 8 + wid;  // B*(T/16) = 2048 tiles
  int b  = tile >> 9;
  int t0 = (tile & 511) << 4;
  int m  = lane & 15;               // A row = token, B col = expert (same id)
  int kh = (lane >> 4) << 1;        // k-pair: lanes 0-15 -> K0,1 ; 16-31 -> K2,3
  bool ev = m < E_;

  const float* ap = X + ((size_t)b * T_ + t0 + m) * D_ + kh;
  const float* bp = &sw[(ev ? m : 8) * 1028 + kh];  // pad lanes -> zero row

  v8f acc = {};
  for (int ko = 0; ko < D_; ko += 64) {
    __builtin_prefetch(ap + ko + 256, 0, 0);  // stream 1KB ahead
#pragma unroll
    for (int ki = 0; ki < 64; ki += 4) {
      int kb = ko + ki;
      acc = wmma4(*(const v2f*)(ap + kb), *(const v2f*)(bp + kb), acc);
    }
  }

  if (ev) {  // only expert columns 0..7 are real
    int hi = lane >> 4;
#pragma unroll
    for (int r = 0; r < 8; ++r) {
      int M = r + (hi << 3);
      logits[((size_t)b * T_ + t0 + M) * E_ + m] = acc[r];
    }
  }
}

// ---------------------------------------------------------------------------
// Kernel 2: dispatch-softmax stats over T: max and 1/sum(exp) per (b,e)
// grid: 32 blocks x 256 thr
// ---------------------------------------------------------------------------
__global__ __launch_bounds__(256) void k_stats(float* __restrict__ ws) {
  const float* logits = ws + OFF_LOGITS;
  float* maxv = ws + OFF_MAXV;
  float* rsum = ws + OFF_RSUM;
  int b = blockIdx.x >> 3, e = blockIdx.x & 7;
  __shared__ float red[256];

  float m = -1e30f;
  for (int t = threadIdx.x; t < T_; t += 256)
    m = fmaxf(m, logits[((size_t)b * T_ + t) * E_ + e]);
  red[threadIdx.x] = m;
  __syncthreads();
  for (int s = 128; s > 0; s >>= 1) {
    if (threadIdx.x < s) red[threadIdx.x] = fmaxf(red[threadIdx.x], red[threadIdx.x + s]);
    __syncthreads();
  }
  float mx = red[0];
  __syncthreads();

  float sum = 0.0f;
  for (int t = threadIdx.x; t < T_; t += 256)
    sum += __expf(logits[((size_t)b * T_ + t) * E_ + e] - mx);
  red[threadIdx.x] = sum;
  __syncthreads();
  for (int s = 128; s > 0; s >>= 1) {
    if (threadIdx.x < s) red[threadIdx.x] += red[threadIdx.x + s];
    __syncthreads();
  }
  if (threadIdx.x == 0) {
    maxv[blockIdx.x] = mx;
    rsum[blockIdx.x] = 1.0f / red[0];
  }
}

// ---------------------------------------------------------------------------
// Kernel 3: slots[b,e,d] = sum_t dispatch_w[b,t,e] * X[b,t,d]
// WMMA: M=E(pad16), N=16 d, K over T (split 16 ways, atomic accumulate)
// Pad lanes get exact zeros via rs=0 (exp(l-max) <= 1, no overflow possible).
// grid: 512 blocks x 256 thr (4096 waves)
// ---------------------------------------------------------------------------
__global__ __launch_bounds__(256) void k_slots(const float* __restrict__ X,
                                               float* __restrict__ ws) {
  const float* logits = ws + OFF_LOGITS;
  const float* maxv = ws + OFF_MAXV;
  const float* rsum = ws + OFF_RSUM;
  float* slots = ws + OFF_SLOTS;

  int wid = threadIdx.x >> 5, lane = threadIdx.x & 31;
  int g = blockIdx.x * 8 + wid;     // B*64*16 = 4096 waves
  int b = g >> 10;
  int rem = g & 1023;
  int d0 = (rem >> 4) << 4;
  int tb0 = (rem & 15) << 9;        // 512-token K split
  int m = lane & 15;                // expert row / d col
  int me8 = m & 7;                  // clamped expert for always-valid addresses
  int kh = (lane >> 4) << 1;
  bool ev = m < E_;

  float me = maxv[b * 8 + me8];
  float rs = ev ? rsum[b * 8 + me8] : 0.0f;  // pad lanes: weight == exact 0

  v8f acc = {};
  for (int tc = tb0; tc < tb0 + 512; tc += 64) {
    if (tc + 64 < tb0 + 512)
      __builtin_prefetch(&X[((size_t)b * T_ + tc + 64) * D_ + d0 + m], 0, 0);
#pragma unroll
    for (int t = tc; t < tc + 64; t += 4) {
      int k0 = t + kh;
      float l0 = logits[((size_t)b * T_ + k0) * E_ + me8];
      float l1 = logits[((size_t)b * T_ + k0 + 1) * E_ + me8];
      v2f av;
      av.x = __expf(l0 - me) * rs;
      av.y = __expf(l1 - me) * rs;
      v2f bv;
      bv.x = X[((size_t)b * T_ + k0) * D_ + d0 + m];
      bv.y = X[((size_t)b * T_ + k0 + 1) * D_ + d0 + m];
      acc = wmma4(av, bv, acc);
    }
  }

  int hi = lane >> 4;
#pragma unroll
  for (int r = 0; r < 8; ++r) {
    int M = r + (hi << 3);
    if (M < E_) atomicAdd(&slots[((size_t)(b * E_ + M)) * D_ + d0 + m], acc[r]);
  }
}

// ---------------------------------------------------------------------------
// Kernel 4: h[b,e,f] = silu(slots.w1) * (slots.w3)   (M=B pad16, N=16 f, K=D)
// grid: 256 blocks x 256 thr (2048 waves, one 16-wide F tile each)
// ---------------------------------------------------------------------------
__global__ __launch_bounds__(256) void k_ffn1(const float* __restrict__ W1,
                                              const float* __restrict__ W3,
                                              float* __restrict__ ws) {
  const float* slots = ws + OFF_SLOTS;
  float* h = ws + OFF_H;
  __shared__ float sl[5 * 1028];  // rows 0-3 = slots, row 4 = zeros

  int e = blockIdx.x >> 5;  // 32 blocks per expert (block-uniform)
  for (int i = threadIdx.x; i < 4 * D_; i += 256) {
    int bb = i >> 10, k = i & 1023;
    sl[bb * 1028 + k] = slots[((size_t)(bb * E_ + e)) * D_ + k];
  }
  for (int i = threadIdx.x; i < 1028; i += 256) sl[4 * 1028 + i] = 0.0f;
  __syncthreads();

  int wid = threadIdx.x >> 5, lane = threadIdx.x & 31;
  int f0 = ((blockIdx.x * 8 + wid) & 255) << 4;
  int m = lane & 15;
  int kh = (lane >> 4) << 1;
  bool mv = m < B_;

  const float* ap  = &sl[(mv ? m : 4) * 1028 + kh];  // pad lanes -> zero row
  const float* b1p = W1 + ((size_t)e * F_ + f0 + m) * D_ + kh;
  const float* b3p = W3 + ((size_t)e * F_ + f0 + m) * D_ + kh;

  v8f acc1 = {}, acc3 = {};
  for (int ko = 0; ko < D_; ko += 64) {
    __builtin_prefetch(b1p + ko + 256, 0, 0);  // weight streams 1KB ahead
    __builtin_prefetch(b3p + ko + 256, 0, 0);
#pragma unroll
    for (int ki = 0; ki < 64; ki += 4) {
      int kb = ko + ki;
      v2f av = *(const v2f*)(ap + kb);
      acc1 = wmma4(av, *(const v2f*)(b1p + kb), acc1);
      acc3 = wmma4(av, *(const v2f*)(b3p + kb), acc3);
    }
  }

  int hi = lane >> 4;
#pragma unroll
  for (int r = 0; r < 8; ++r) {
    int M = r + (hi << 3);
    if (M < B_) {
      float x1 = acc1[r];
      float hv = (x1 / (1.0f + __expf(-x1))) * acc3[r];  // silu(x1)*x3
      h[((size_t)(M * E_ + e)) * F_ + f0 + m] = hv;
    }
  }
}

// ---------------------------------------------------------------------------
// Kernel 5: y[b,e,d] = sum_f h[b,e,f] * w2[e,d,f]   (K=F split 4 ways, atomics)
// grid: 256 blocks x 256 thr (2048 waves)
// ---------------------------------------------------------------------------
__global__ __launch_bounds__(256) void k_ffn2(const float* __restrict__ W2,
                                              float* __restrict__ ws) {
  const float* h = ws + OFF_H;
  float* y = ws + OFF_Y;
  __shared__ float sl[5 * 1028];  // rows 0-3 = h chunk, row 4 = zeros

  int g0 = blockIdx.x * 8;
  int e  = g0 >> 8;                     // block-uniform expert
  int fb0 = (((g0 & 255) >> 6) << 10);  // block-uniform 1024-wide K split
  for (int i = threadIdx.x; i < 4 * 1024; i += 256) {
    int bb = i >> 10, k = i & 1023;
    sl[bb * 1028 + k] = h[((size_t)(bb * E_ + e)) * F_ + fb0 + k];
  }
  for (int i = threadIdx.x; i < 1028; i += 256) sl[4 * 1028 + i] = 0.0f;
  __syncthreads();

  int wid = threadIdx.x >> 5, lane = threadIdx.x & 31;
  int d0 = ((blockIdx.x * 8 + wid) & 63) << 4;
  int m = lane & 15;
  int kh = (lane >> 4) << 1;
  bool mv = m < B_;

  const float* ap = &sl[(mv ? m : 4) * 1028 + kh];  // pad lanes -> zero row
  const float* bp = W2 + ((size_t)e * D_ + d0 + m) * F_ + fb0 + kh;

  v8f acc = {};
  for (int ko = 0; ko < 1024; ko += 64) {
    __builtin_prefetch(bp + ko + 256, 0, 0);
#pragma unroll
    for (int ki = 0; ki < 64; ki += 4) {
      int kb = ko + ki;
      acc = wmma4(*(const v2f*)(ap + kb), *(const v2f*)(bp + kb), acc);
    }
  }

  int hi = lane >> 4;
#pragma unroll
  for (int r = 0; r < 8; ++r) {
    int M = r + (hi << 3);
    if (M < B_) atomicAdd(&y[((size_t)(M * E_ + e)) * D_ + d0 + m], acc[r]);
  }
}

// ---------------------------------------------------------------------------
// Kernel 6: out[b,t,d] = sum_e softmaxE(logits[b,t,:])[e] * y[b,e,d]
// K=8 -> two WMMAs per 16x16 tile. grid: 16384 blocks x 256 thr
// ---------------------------------------------------------------------------
__global__ __launch_bounds__(256) void k_combine(float* __restrict__ out,
                                                 const float* __restrict__ ws) {
  const float* logits = ws + OFF_LOGITS;
  const float* y = ws + OFF_Y;

  int wid = threadIdx.x >> 5, lane = threadIdx.x & 31;
  int g = blockIdx.x * 8 + wid;     // B*512*64 = 131072 tiles
  int b = g >> 15;
  int rem = g & 32767;
  int t0 = (rem >> 6) << 4;
  int d0 = (rem & 63) << 4;
  int m = lane & 15;
  bool hi = lane >= 16;

  // combine softmax over E=8 for token t0+m (A-operand row)
  const float4* lp = (const float4*)(logits + ((size_t)b * T_ + t0 + m) * E_);
  float4 l0 = lp[0], l1 = lp[1];
  float l[8] = {l0.x, l0.y, l0.z, l0.w, l1.x, l1.y, l1.z, l1.w};
  float mx = l[0];
#pragma unroll
  for (int i = 1; i < 8; ++i) mx = fmaxf(mx, l[i]);
  float w[8], s = 0.0f;
#pragma unroll
  for (int i = 0; i < 8; ++i) { w[i] = __expf(l[i] - mx); s += w[i]; }
  float inv = 1.0f / s;

  v2f a0, a1;
  a0.x = (hi ? w[2] : w[0]) * inv;
  a0.y = (hi ? w[3] : w[1]) * inv;
  a1.x = (hi ? w[6] : w[4]) * inv;
  a1.y = (hi ? w[7] : w[5]) * inv;

  int kb = hi ? 2 : 0;
  const float* yb = y + (size_t)b * E_ * D_ + d0 + m;
  v2f b0, b1;
  b0.x = yb[(size_t)(kb + 0) * D_];
  b0.y = yb[(size_t)(kb + 1) * D_];
  b1.x = yb[(size_t)(kb + 4) * D_];
  b1.y = yb[(size_t)(kb + 5) * D_];

  v8f acc = {};
  acc = wmma4(a0, b0, acc);
  acc = wmma4(a1, b1, acc);

#pragma unroll
  for (int r = 0; r < 8; ++r) {
    int M = r + ((lane >> 4) << 3);
    out[((size_t)b * T_ + t0 + M) * D_ + d0 + m] = acc[r];
  }
}

// ---------------------------------------------------------------------------
extern "C" void kernel_launch(void* const* d_in, const int* in_sizes, int n_in,
                              void* d_out, int out_size, void* d_ws, size_t ws_size,
                              hipStream_t stream) {
  (void)in_sizes; (void)n_in; (void)out_size; (void)ws_size;
  const float* X  = (const float*)d_in[0];
  const float* SW = (const float*)d_in[1];
  const float* W1 = (const float*)d_in[2];
  const float* W3 = (const float*)d_in[3];
  const float* W2 = (const float*)d_in[4];
  float* out = (float*)d_out;
  float* ws  = (float*)d_ws;

  k_init   <<<768,   256, 0, stream>>>(ws);
  k_logits <<<256,   256, 0, stream>>>(X, SW, ws);
  k_stats  <<<32,    256, 0, stream>>>(ws);
  k_slots  <<<512,   256, 0, stream>>>(X, ws);
  k_ffn1   <<<256,   256, 0, stream>>>(W1, W3, ws);
  k_ffn2   <<<256,   256, 0, stream>>>(W2, ws);
  k_combine<<<16384, 256, 0, stream>>>(out, ws);
}